// WindowLayer_74199855005991
// MI455X (gfx1250) — compile-verified
//
#include <hip/hip_runtime.h>
#include <hip/hip_bf16.h>

#define HIDDEN 1024
#define NGAUSS 20
#define BATCH  512
#define SEQ    1024
#define VOCAB  128

typedef float v2f __attribute__((ext_vector_type(2)));
typedef float v8f __attribute__((ext_vector_type(8)));

// ---------------------------------------------------------------------------
// Kernel 1: lin = X @ W^T + b, fused exp activations.
// D = A x B (+acc) with V_WMMA_F32_16X16X4_F32.
//   A (16x4, MxK): lane L -> M = L&15, VGPR0 holds K = (L>>4)*2, VGPR1 K+1
//   B (4x16, KxN): lane L -> N = L&15, VGPR0 holds K = (L>>4)*2, VGPR1 K+1
//   C/D (16x16):   VGPR r: lanes 0-15 M=r, lanes 16-31 M=r+8; N = lane&15
// Grid: 32 blocks (row tiles of 16), 128 threads = 4 waves (col tiles of 16,
// covering N=0..63; N=60..63 are clamped reads, masked on writeback).
// ---------------------------------------------------------------------------
__global__ __launch_bounds__(128) void window_linear_wmma(
    const float* __restrict__ X,          // [512,1024]
    const float* __restrict__ W,          // [60,1024]
    const float* __restrict__ bias,       // [60]
    const float* __restrict__ kappa_prev, // [512,20]
    float* __restrict__ alpha_ws,         // [512,20]
    float* __restrict__ beta_ws,          // [512,20]
    float* __restrict__ kappa_ws,         // [512,20]
    float* __restrict__ kappa_out)        // [512,20] (d_out slice)
{
    const int lane    = threadIdx.x & 31;
    const int wave    = threadIdx.x >> 5;      // col tile 0..3
    const int rowBase = blockIdx.x * 16;
    const int colBase = wave * 16;

    const int mA    = lane & 15;
    const int kHalf = (lane >> 4) * 2;         // 0 or 2 (K sub-index)
    const int nB    = colBase + (lane & 15);
    const int nBc   = nB < 60 ? nB : 59;       // clamp pad cols (masked later)

    const float* arow = X + (size_t)(rowBase + mA) * HIDDEN + kHalf;
    const float* brow = W + (size_t)nBc * HIDDEN + kHalf;

    v8f acc = {};
#pragma unroll 4
    for (int k0 = 0; k0 < HIDDEN; k0 += 4) {
        v2f a  = *(const v2f*)(arow + k0);
        v2f bb = *(const v2f*)(brow + k0);
        acc = __builtin_amdgcn_wmma_f32_16x16x4_f32(
            /*neg_a=*/false, a, /*neg_b=*/false, bb,
            /*c_mod=*/(short)0, acc, /*reuse_a=*/false, /*reuse_b=*/false);
    }

    // Writeback with fused activations (EXEC divergence only after all WMMAs).
    const int ng  = colBase + (lane & 15);
    const int mHi = (lane >> 4) * 8;
    if (ng < 60) {
        const float bv = bias[ng];
#pragma unroll
        for (int r = 0; r < 8; ++r) {
            const int mg  = rowBase + mHi + r;
            const float lin = acc[r] + bv;
            if (ng < NGAUSS) {
                alpha_ws[mg * NGAUSS + ng] = __expf(lin);
            } else if (ng < 2 * NGAUSS) {
                beta_ws[mg * NGAUSS + (ng - NGAUSS)] = __expf(lin);
            } else {
                const int kk = ng - 2 * NGAUSS;
                const float kp = kappa_prev[mg * NGAUSS + kk] + __expf(lin) * 0.05f;
                kappa_ws[mg * NGAUSS + kk]  = kp;
                kappa_out[mg * NGAUSS + kk] = kp;
            }
        }
    }
}

// ---------------------------------------------------------------------------
// Kernel 2: phi[b,u] = sum_k alpha*exp(-beta*(kappa-u)^2).  One block per b.
// ---------------------------------------------------------------------------
__global__ __launch_bounds__(256) void window_phi_kernel(
    const float* __restrict__ alpha,
    const float* __restrict__ beta,
    const float* __restrict__ kappa,
    float* __restrict__ phi_out)          // [512,1024] (d_out slice)
{
    __shared__ float al[NGAUSS], be[NGAUSS], ka[NGAUSS];
    const int b = blockIdx.x;
    const int t = threadIdx.x;
    if (t < NGAUSS) {
        al[t] = alpha[b * NGAUSS + t];
        be[t] = beta [b * NGAUSS + t];
        ka[t] = kappa[b * NGAUSS + t];
    }
    __syncthreads();

#pragma unroll
    for (int uo = 0; uo < SEQ / 256; ++uo) {
        const int u = uo * 256 + t;
        const float uf = (float)u;
        float s = 0.0f;
#pragma unroll
        for (int k = 0; k < NGAUSS; ++k) {
            const float d = ka[k] - uf;
            s += al[k] * __expf(-be[k] * d * d);
        }
        phi_out[(size_t)b * SEQ + u] = s;
    }
}

// ---------------------------------------------------------------------------
// Kernel 3: wind[b,v] = sum_u phi[b,u] * sent[b,u,v].  Bandwidth-bound:
// reads 268 MB of sentence_hot. One block (8 waves) per batch; lane owns 4
// vocab cols (float4 -> global_load_b128, fully coalesced 512B per u-row);
// waves split U; LDS reduction at the end.
// ---------------------------------------------------------------------------
__global__ __launch_bounds__(256) void window_wind_kernel(
    const float* __restrict__ sent,       // [512,1024,128]
    const float* __restrict__ phi,        // [512,1024]
    float* __restrict__ wind)             // [512,128] (d_out slice)
{
    __shared__ float  phs[SEQ];           // 4 KB
    __shared__ float4 red[256];           // 4 KB

    const int b = blockIdx.x;
    const int t = threadIdx.x;

    for (int i = t; i < SEQ; i += 256)
        phs[i] = phi[(size_t)b * SEQ + i];
    __syncthreads();

    const int lane = t & 31;              // vocab group: cols lane*4..lane*4+3
    const int wv   = t >> 5;              // u-slice 0..7 (128 u each)
    const float4* row = (const float4*)(sent + (size_t)b * SEQ * VOCAB);

    float4 acc = make_float4(0.f, 0.f, 0.f, 0.f);
#pragma unroll 4
    for (int uu = 0; uu < SEQ / 8; ++uu) {
        const int u = wv * (SEQ / 8) + uu;
        const float4 s = row[(size_t)u * (VOCAB / 4) + lane];
        const float  p = phs[u];
        acc.x += p * s.x; acc.y += p * s.y;
        acc.z += p * s.z; acc.w += p * s.w;
    }
    red[t] = acc;
    __syncthreads();

    if (t < 32) {
        float4 r = red[t];
#pragma unroll
        for (int w = 1; w < 8; ++w) {
            const float4 o = red[w * 32 + t];
            r.x += o.x; r.y += o.y; r.z += o.z; r.w += o.w;
        }
        ((float4*)(wind + (size_t)b * VOCAB))[t] = r;
    }
}

// ---------------------------------------------------------------------------
extern "C" void kernel_launch(void* const* d_in, const int* in_sizes, int n_in,
                              void* d_out, int out_size, void* d_ws, size_t ws_size,
                              hipStream_t stream) {
    const float* X     = (const float*)d_in[0];  // window_input [512,1024]
    const float* sent  = (const float*)d_in[1];  // sentence_hot [512,1024,128]
    const float* kprev = (const float*)d_in[2];  // kappa_prev   [512,20]
    const float* W     = (const float*)d_in[3];  // W            [60,1024]
    const float* bias  = (const float*)d_in[4];  // b            [60]

    float* out = (float*)d_out;
    float* wind_out  = out;                              // 512*128 = 65536
    float* kappa_out = out + BATCH * VOCAB;              // 512*20  = 10240
    float* phi_out   = out + BATCH * VOCAB + BATCH * NGAUSS; // 512*1024

    // Workspace: alpha/beta/kappa, 3 * 512*20 floats = 120 KB.
    float* alpha_ws = (float*)d_ws;
    float* beta_ws  = alpha_ws + BATCH * NGAUSS;
    float* kappa_ws = beta_ws  + BATCH * NGAUSS;

    window_linear_wmma<<<BATCH / 16, 128, 0, stream>>>(
        X, W, bias, kprev, alpha_ws, beta_ws, kappa_ws, kappa_out);

    window_phi_kernel<<<BATCH, 256, 0, stream>>>(
        alpha_ws, beta_ws, kappa_ws, phi_out);

    window_wind_kernel<<<BATCH, 256, 0, stream>>>(
        sent, phi_out, wind_out);
}